// GATModel_82446192214439
// MI455X (gfx1250) — compile-verified
//
#include <hip/hip_runtime.h>
#include <hip/hip_bf16.h>

#define NNODES 4096
#define NFEAT  1024
#define NHID   8
#define NHEADS 8
#define NCLS   16
#define GAT_ALPHA 0.2f

typedef __attribute__((ext_vector_type(2))) float v2f;
typedef __attribute__((ext_vector_type(8))) float v8f;

#if defined(__HIP_DEVICE_COMPILE__) && __has_builtin(__builtin_amdgcn_wmma_f32_16x16x4_f32)
#define HAVE_WMMA_F32 1
#else
#define HAVE_WMMA_F32 0
#endif

// -------------------------------------------------------------------------
// GEMM1: Wh1[4096,64] = x[4096,1024] @ Wcat[1024,64]
// Wcat(k, n) = W_heads[n>>3][k][n&7]  (heads concatenated on N)
// One wave per 16x16 output tile; 4 waves (N-tiles) per block, 256 M-tiles.
// -------------------------------------------------------------------------
__global__ __launch_bounds__(128) void gemm1_wmma(const float* __restrict__ x,
                                                  const float* __restrict__ Wheads,
                                                  float* __restrict__ Wh1) {
  const int lane = threadIdx.x & 31;
  const int wave = threadIdx.x >> 5;
  const int m0 = blockIdx.x * 16;
  const int n0 = wave * 16;
  const int n  = n0 + (lane & 15);
  // B column base: element (k,n) lives at Wheads[(n>>3)*1024*8 + k*8 + (n&7)]
  const float* Bcol = Wheads + (size_t)(n >> 3) * NFEAT * NHID + (n & 7);
#if HAVE_WMMA_F32
  const int kp = lane >> 4;                 // which K-pair half this lane holds
  const float* Arow = x + (size_t)(m0 + (lane & 15)) * NFEAT + 2 * kp;
  v8f c = {0.f,0.f,0.f,0.f,0.f,0.f,0.f,0.f};
  #pragma unroll 4
  for (int k0 = 0; k0 < NFEAT; k0 += 4) {
    v2f a = *(const v2f*)(Arow + k0);       // A[m][k..k+1], K-halves split by lane group
    const int k = k0 + 2 * kp;
    v2f b;
    b.x = Bcol[(size_t)k * NHID];
    b.y = Bcol[(size_t)(k + 1) * NHID];
    c = __builtin_amdgcn_wmma_f32_16x16x4_f32(false, a, false, b, (short)0, c,
                                              false, false);
  }
  const int mrow = m0 + (lane >> 4) * 8;    // C/D layout: vgpr v -> M = v (+8 for hi lanes)
  #pragma unroll
  for (int v = 0; v < 8; ++v)
    Wh1[(size_t)(mrow + v) * (NHEADS * NHID) + n] = c[v];
#else
  const int mrow = m0 + (lane >> 4) * 8;
  #pragma unroll
  for (int v = 0; v < 8; ++v) {
    const float* ar = x + (size_t)(mrow + v) * NFEAT;
    float s = 0.f;
    for (int k = 0; k < NFEAT; ++k) s = fmaf(ar[k], Bcol[(size_t)k * NHID], s);
    Wh1[(size_t)(mrow + v) * (NHEADS * NHID) + n] = s;
  }
#endif
}

// -------------------------------------------------------------------------
// GEMM2: Wh2[4096,16] = h2[4096,64] @ W_final[64,16]   (one wave per block)
// -------------------------------------------------------------------------
__global__ __launch_bounds__(32) void gemm2_wmma(const float* __restrict__ h2,
                                                 const float* __restrict__ Wfin,
                                                 float* __restrict__ Wh2) {
  const int lane = threadIdx.x & 31;
  const int m0 = blockIdx.x * 16;
  const int n  = lane & 15;
#if HAVE_WMMA_F32
  const int kp = lane >> 4;
  const float* Arow = h2 + (size_t)(m0 + (lane & 15)) * 64 + 2 * kp;
  v8f c = {0.f,0.f,0.f,0.f,0.f,0.f,0.f,0.f};
  #pragma unroll
  for (int k0 = 0; k0 < 64; k0 += 4) {
    v2f a = *(const v2f*)(Arow + k0);
    const int k = k0 + 2 * kp;
    v2f b;
    b.x = Wfin[k * NCLS + n];
    b.y = Wfin[(k + 1) * NCLS + n];
    c = __builtin_amdgcn_wmma_f32_16x16x4_f32(false, a, false, b, (short)0, c,
                                              false, false);
  }
  const int mrow = m0 + (lane >> 4) * 8;
  #pragma unroll
  for (int v = 0; v < 8; ++v)
    Wh2[(size_t)(mrow + v) * NCLS + n] = c[v];
#else
  const int mrow = m0 + (lane >> 4) * 8;
  #pragma unroll
  for (int v = 0; v < 8; ++v) {
    const float* ar = h2 + (size_t)(mrow + v) * 64;
    float s = 0.f;
    for (int k = 0; k < 64; ++k) s = fmaf(ar[k], Wfin[k * NCLS + n], s);
    Wh2[(size_t)(mrow + v) * NCLS + n] = s;
  }
#endif
}

// -------------------------------------------------------------------------
// Per-node attention features, layer 1: fsrc/fdst[i][h] = Wh[i,h*8:+8] . a[h][:8 / 8:]
// -------------------------------------------------------------------------
__global__ void feat1(const float* __restrict__ Wh1, const float* __restrict__ aheads,
                      float* __restrict__ fsrc, float* __restrict__ fdst) {
  const int i = blockIdx.x * blockDim.x + threadIdx.x;
  if (i >= NNODES) return;
  const float* wh = Wh1 + (size_t)i * (NHEADS * NHID);
  #pragma unroll
  for (int h = 0; h < NHEADS; ++h) {
    const float* a = aheads + h * (2 * NHID);
    float s = 0.f, d = 0.f;
    #pragma unroll
    for (int o = 0; o < NHID; ++o) {
      const float w = wh[h * NHID + o];
      s = fmaf(w, a[o], s);
      d = fmaf(w, a[NHID + o], d);
    }
    fsrc[(size_t)i * NHEADS + h] = s;
    fdst[(size_t)i * NHEADS + h] = d;
  }
}

__global__ void feat2(const float* __restrict__ Wh2, const float* __restrict__ afin,
                      float* __restrict__ fsrc2, float* __restrict__ fdst2) {
  const int i = blockIdx.x * blockDim.x + threadIdx.x;
  if (i >= NNODES) return;
  const float* wh = Wh2 + (size_t)i * NCLS;
  float s = 0.f, d = 0.f;
  #pragma unroll
  for (int c = 0; c < NCLS; ++c) {
    s = fmaf(wh[c], afin[c], s);
    d = fmaf(wh[c], afin[NCLS + c], d);
  }
  fsrc2[i] = s;
  fdst2[i] = d;
}

// -------------------------------------------------------------------------
// Fused masked softmax attention, layer 1 (8 heads, Fout=8), one wave per row.
// Pass 1: per-head max of fdst over neighbors (lrelu is monotone, so
//         max_j lrelu(fs+fd_j) = lrelu(fs + max_j fd_j)).
// Pass 2: exp-sum + weighted accumulation of Wh rows. ELU epilogue.
// -------------------------------------------------------------------------
__global__ __launch_bounds__(32) void attn1(const float* __restrict__ adj,
                                            const float* __restrict__ Wh1,
                                            const float* __restrict__ fsrc,
                                            const float* __restrict__ fdst,
                                            float* __restrict__ h2out) {
  const int i = blockIdx.x;
  const int lane = threadIdx.x;
  const float* arow = adj + (size_t)i * NNODES;

  float fs[8];
  {
    const float4* p = (const float4*)(fsrc + (size_t)i * NHEADS);
    const float4 a0 = p[0], a1 = p[1];
    fs[0]=a0.x; fs[1]=a0.y; fs[2]=a0.z; fs[3]=a0.w;
    fs[4]=a1.x; fs[5]=a1.y; fs[6]=a1.z; fs[7]=a1.w;
  }

  // ---- pass 1: neighbor max of fdst per head
  float mfd[8];
  #pragma unroll
  for (int h = 0; h < 8; ++h) mfd[h] = -3.0e38f;
  auto p1 = [&](int j) {
    const float4* p = (const float4*)(fdst + (size_t)j * NHEADS);
    const float4 a0 = p[0], a1 = p[1];
    mfd[0]=fmaxf(mfd[0],a0.x); mfd[1]=fmaxf(mfd[1],a0.y);
    mfd[2]=fmaxf(mfd[2],a0.z); mfd[3]=fmaxf(mfd[3],a0.w);
    mfd[4]=fmaxf(mfd[4],a1.x); mfd[5]=fmaxf(mfd[5],a1.y);
    mfd[6]=fmaxf(mfd[6],a1.z); mfd[7]=fmaxf(mfd[7],a1.w);
  };
  for (int it = 0; it < NNODES / 128; ++it) {
    const int j0 = it * 128 + lane * 4;
    const float4 av = *(const float4*)(arow + j0);
    if (av.x > 0.f) p1(j0 + 0);
    if (av.y > 0.f) p1(j0 + 1);
    if (av.z > 0.f) p1(j0 + 2);
    if (av.w > 0.f) p1(j0 + 3);
  }
  #pragma unroll
  for (int mask = 16; mask >= 1; mask >>= 1) {
    #pragma unroll
    for (int h = 0; h < 8; ++h) mfd[h] = fmaxf(mfd[h], __shfl_xor(mfd[h], mask, 32));
  }
  float mh[8];
  #pragma unroll
  for (int h = 0; h < 8; ++h) {
    const float e = fs[h] + mfd[h];
    mh[h] = (e > 0.f) ? e : GAT_ALPHA * e;
  }

  // ---- pass 2: softmax-weighted accumulation
  float l8[8];
  float acc[64];
  #pragma unroll
  for (int h = 0; h < 8; ++h) l8[h] = 0.f;
  #pragma unroll
  for (int c = 0; c < 64; ++c) acc[c] = 0.f;
  auto p2 = [&](int j) {
    const float4* dp = (const float4*)(fdst + (size_t)j * NHEADS);
    const float4 d0 = dp[0], d1 = dp[1];
    const float fd[8] = {d0.x,d0.y,d0.z,d0.w,d1.x,d1.y,d1.z,d1.w};
    const float* whj = Wh1 + (size_t)j * (NHEADS * NHID);
    #pragma unroll
    for (int h = 0; h < 8; ++h) {
      float e = fs[h] + fd[h];
      e = (e > 0.f) ? e : GAT_ALPHA * e;
      const float pw = __expf(e - mh[h]);
      l8[h] += pw;
      const float4 w0 = *(const float4*)(whj + h * 8);
      const float4 w1 = *(const float4*)(whj + h * 8 + 4);
      acc[h*8+0] = fmaf(pw, w0.x, acc[h*8+0]);
      acc[h*8+1] = fmaf(pw, w0.y, acc[h*8+1]);
      acc[h*8+2] = fmaf(pw, w0.z, acc[h*8+2]);
      acc[h*8+3] = fmaf(pw, w0.w, acc[h*8+3]);
      acc[h*8+4] = fmaf(pw, w1.x, acc[h*8+4]);
      acc[h*8+5] = fmaf(pw, w1.y, acc[h*8+5]);
      acc[h*8+6] = fmaf(pw, w1.z, acc[h*8+6]);
      acc[h*8+7] = fmaf(pw, w1.w, acc[h*8+7]);
    }
  };
  for (int it = 0; it < NNODES / 128; ++it) {
    const int j0 = it * 128 + lane * 4;
    const float4 av = *(const float4*)(arow + j0);
    if (av.x > 0.f) p2(j0 + 0);
    if (av.y > 0.f) p2(j0 + 1);
    if (av.z > 0.f) p2(j0 + 2);
    if (av.w > 0.f) p2(j0 + 3);
  }
  // wave32 butterfly reduction
  #pragma unroll
  for (int mask = 16; mask >= 1; mask >>= 1) {
    #pragma unroll
    for (int h = 0; h < 8; ++h) l8[h] += __shfl_xor(l8[h], mask, 32);
    #pragma unroll
    for (int c = 0; c < 64; ++c) acc[c] += __shfl_xor(acc[c], mask, 32);
  }
  if (lane == 0) {
    #pragma unroll
    for (int c = 0; c < 64; ++c) {
      float v = acc[c] / l8[c >> 3];
      v = (v > 0.f) ? v : (__expf(v) - 1.f);   // ELU
      h2out[(size_t)i * 64 + c] = v;
    }
  }
}

// -------------------------------------------------------------------------
// Fused masked softmax attention, layer 2 (1 "head", Fout=16) + log_softmax.
// -------------------------------------------------------------------------
__global__ __launch_bounds__(32) void attn2(const float* __restrict__ adj,
                                            const float* __restrict__ Wh2,
                                            const float* __restrict__ fsrc2,
                                            const float* __restrict__ fdst2,
                                            float* __restrict__ out) {
  const int i = blockIdx.x;
  const int lane = threadIdx.x;
  const float* arow = adj + (size_t)i * NNODES;
  const float fs = fsrc2[i];

  float mfd = -3.0e38f;
  for (int it = 0; it < NNODES / 128; ++it) {
    const int j0 = it * 128 + lane * 4;
    const float4 av = *(const float4*)(arow + j0);
    if (av.x > 0.f) mfd = fmaxf(mfd, fdst2[j0 + 0]);
    if (av.y > 0.f) mfd = fmaxf(mfd, fdst2[j0 + 1]);
    if (av.z > 0.f) mfd = fmaxf(mfd, fdst2[j0 + 2]);
    if (av.w > 0.f) mfd = fmaxf(mfd, fdst2[j0 + 3]);
  }
  #pragma unroll
  for (int mask = 16; mask >= 1; mask >>= 1)
    mfd = fmaxf(mfd, __shfl_xor(mfd, mask, 32));
  const float e0 = fs + mfd;
  const float mh = (e0 > 0.f) ? e0 : GAT_ALPHA * e0;

  float l = 0.f;
  float acc[16];
  #pragma unroll
  for (int c = 0; c < 16; ++c) acc[c] = 0.f;
  auto p2 = [&](int j) {
    float e = fs + fdst2[j];
    e = (e > 0.f) ? e : GAT_ALPHA * e;
    const float pw = __expf(e - mh);
    l += pw;
    const float4* wp = (const float4*)(Wh2 + (size_t)j * NCLS);
    const float4 w0 = wp[0], w1 = wp[1], w2 = wp[2], w3 = wp[3];
    acc[0]  = fmaf(pw, w0.x, acc[0]);  acc[1]  = fmaf(pw, w0.y, acc[1]);
    acc[2]  = fmaf(pw, w0.z, acc[2]);  acc[3]  = fmaf(pw, w0.w, acc[3]);
    acc[4]  = fmaf(pw, w1.x, acc[4]);  acc[5]  = fmaf(pw, w1.y, acc[5]);
    acc[6]  = fmaf(pw, w1.z, acc[6]);  acc[7]  = fmaf(pw, w1.w, acc[7]);
    acc[8]  = fmaf(pw, w2.x, acc[8]);  acc[9]  = fmaf(pw, w2.y, acc[9]);
    acc[10] = fmaf(pw, w2.z, acc[10]); acc[11] = fmaf(pw, w2.w, acc[11]);
    acc[12] = fmaf(pw, w3.x, acc[12]); acc[13] = fmaf(pw, w3.y, acc[13]);
    acc[14] = fmaf(pw, w3.z, acc[14]); acc[15] = fmaf(pw, w3.w, acc[15]);
  };
  for (int it = 0; it < NNODES / 128; ++it) {
    const int j0 = it * 128 + lane * 4;
    const float4 av = *(const float4*)(arow + j0);
    if (av.x > 0.f) p2(j0 + 0);
    if (av.y > 0.f) p2(j0 + 1);
    if (av.z > 0.f) p2(j0 + 2);
    if (av.w > 0.f) p2(j0 + 3);
  }
  #pragma unroll
  for (int mask = 16; mask >= 1; mask >>= 1) {
    l += __shfl_xor(l, mask, 32);
    #pragma unroll
    for (int c = 0; c < 16; ++c) acc[c] += __shfl_xor(acc[c], mask, 32);
  }
  if (lane == 0) {
    float o[16], mx = -3.0e38f;
    #pragma unroll
    for (int c = 0; c < 16; ++c) { o[c] = acc[c] / l; mx = fmaxf(mx, o[c]); }
    float s = 0.f;
    #pragma unroll
    for (int c = 0; c < 16; ++c) s += __expf(o[c] - mx);
    const float lse = mx + __logf(s);
    #pragma unroll
    for (int c = 0; c < 16; ++c) out[(size_t)i * NCLS + c] = o[c] - lse;
  }
}

// -------------------------------------------------------------------------
extern "C" void kernel_launch(void* const* d_in, const int* in_sizes, int n_in,
                              void* d_out, int out_size, void* d_ws, size_t ws_size,
                              hipStream_t stream) {
  const float* x      = (const float*)d_in[0];  // [4096,1024]
  const float* adj    = (const float*)d_in[1];  // [4096,4096]
  const float* Wheads = (const float*)d_in[2];  // [8,1024,8]
  const float* aheads = (const float*)d_in[3];  // [8,16,1]
  const float* Wfin   = (const float*)d_in[4];  // [64,16]
  const float* afin   = (const float*)d_in[5];  // [32,1]
  float* out = (float*)d_out;                   // [4096,16]

  float* ws    = (float*)d_ws;
  float* Wh1   = ws;                             // 4096*64
  float* fsrc1 = Wh1 + (size_t)NNODES * 64;      // 4096*8
  float* fdst1 = fsrc1 + (size_t)NNODES * 8;     // 4096*8
  float* h2    = fdst1 + (size_t)NNODES * 8;     // 4096*64
  float* Wh2   = h2 + (size_t)NNODES * 64;       // 4096*16
  float* fsrc2 = Wh2 + (size_t)NNODES * 16;      // 4096
  float* fdst2 = fsrc2 + (size_t)NNODES;         // 4096

  gemm1_wmma<<<NNODES / 16, 128, 0, stream>>>(x, Wheads, Wh1);
  feat1<<<NNODES / 256, 256, 0, stream>>>(Wh1, aheads, fsrc1, fdst1);
  attn1<<<NNODES, 32, 0, stream>>>(adj, Wh1, fsrc1, fdst1, h2);
  gemm2_wmma<<<NNODES / 16, 32, 0, stream>>>(h2, Wfin, Wh2);
  feat2<<<NNODES / 256, 256, 0, stream>>>(Wh2, afin, fsrc2, fdst2);
  attn2<<<NNODES, 32, 0, stream>>>(adj, Wh2, fsrc2, fdst2, out);

  (void)in_sizes; (void)n_in; (void)out_size; (void)ws_size;
}